// ChordalPCWeightTransform_10591389352174
// MI455X (gfx1250) — compile-verified
//
#include <hip/hip_runtime.h>
#include <stdint.h>

// out[b,l,j] = softmax_j( x[b,l,j] * w[(j - root(l)) mod 12] ), j=12 uses w[12].
// (The two torch.roll gathers in the reference compose to the identity; only the
//  rotated weight survives.)  Pure HBM-streaming kernel: ~0.98 GB traffic -> ~42us
//  floor at 23.3 TB/s.  Data path: CDNA5 async global<->LDS DMA for coalescing of
//  the 52-byte rows; LDS stride-13 row reads are bank-conflict free (gcd(13,64)=1).
// Single-pass stream (~5x the 192MB L2) -> NT temporal hints on both directions
// to avoid L2 churn.

#define P               13
#define BLOCK           256
#define ROWS_PER_BLOCK  256
#define ELEMS_PER_BLOCK (ROWS_PER_BLOCK * P)   // 3328 floats = 13312 B

__global__ __launch_bounds__(BLOCK) void chordal_pc_softmax_kernel(
    const float* __restrict__ x,      // [B*L*P] contiguous
    const float* __restrict__ w,      // [13]
    float* __restrict__ out)          // [B*L*P]
{
    __shared__ float sRow[ELEMS_PER_BLOCK];
    __shared__ float sW[P];

    const uint32_t t         = threadIdx.x;
    const uint32_t blockElem = blockIdx.x * (uint32_t)ELEMS_PER_BLOCK;

    // Per-lane LDS byte address (group-segment offset = low 32 bits of generic ptr)
    uint32_t lds  = (uint32_t)(uintptr_t)(&sRow[t]);
    // Per-lane global byte offset (GVS mode: saddr = base, vaddr = 32-bit offset).
    uint32_t voff = (blockElem + t) * 4u;

    // Async DMA global -> LDS, 13 x b32 per lane.  INST_OFFSET is added to BOTH the
    // LDS and the global address, so one immediate strides both by 256 floats.
    // NT: lines are touched exactly once -> evict early from GL2.
#define ALOAD(K) asm volatile(                                        \
        "global_load_async_to_lds_b32 %0, %1, %2 offset:" #K          \
        " th:TH_LOAD_NT"                                              \
        :: "v"(lds), "v"(voff), "s"(x) : "memory")
    ALOAD(0);    ALOAD(1024); ALOAD(2048); ALOAD(3072);
    ALOAD(4096); ALOAD(5120); ALOAD(6144); ALOAD(7168);
    ALOAD(8192); ALOAD(9216); ALOAD(10240); ALOAD(11264);
    ALOAD(12288);
#undef ALOAD

    // Stage the 13 weights while the DMA is in flight.
    if (t < P) sW[t] = w[t];

    asm volatile("s_wait_asynccnt 0" ::: "memory");
    __syncthreads();

    // Row owned by this thread; root pitch class r = (l / num_quality) with
    // L = 144, num_quality = 12  =>  r = (row / 12) % 12.
    const uint32_t row = blockIdx.x * (uint32_t)ROWS_PER_BLOCK + t;
    const uint32_t r   = (row / 12u) % 12u;

    float v[P];
#pragma unroll
    for (int j = 0; j < 12; ++j) {
        uint32_t wi = (uint32_t)(j + 12) - r;          // in [1, 23]
        wi = (wi >= 12u) ? wi - 12u : wi;              // (j - r) mod 12
        v[j] = sRow[t * P + j] * sW[wi];
    }
    v[12] = sRow[t * P + 12] * sW[12];

    // 13-wide softmax in registers.
    float m = v[0];
#pragma unroll
    for (int j = 1; j < P; ++j) m = fmaxf(m, v[j]);
    float s = 0.0f;
#pragma unroll
    for (int j = 0; j < P; ++j) { v[j] = __expf(v[j] - m); s += v[j]; }
    const float inv = 1.0f / s;
#pragma unroll
    for (int j = 0; j < P; ++j) sRow[t * P + j] = v[j] * inv;

    __syncthreads();

    // Async DMA LDS -> global; NT store (output never re-read on device).
    // S_ENDPGM implicitly waits all counters.
#define ASTORE(K) asm volatile(                                       \
        "global_store_async_from_lds_b32 %0, %1, %2 offset:" #K       \
        " th:TH_STORE_NT"                                             \
        :: "v"(voff), "v"(lds), "s"(out) : "memory")
    ASTORE(0);    ASTORE(1024); ASTORE(2048); ASTORE(3072);
    ASTORE(4096); ASTORE(5120); ASTORE(6144); ASTORE(7168);
    ASTORE(8192); ASTORE(9216); ASTORE(10240); ASTORE(11264);
    ASTORE(12288);
#undef ASTORE
    asm volatile("s_wait_asynccnt 0" ::: "memory");
}

extern "C" void kernel_launch(void* const* d_in, const int* in_sizes, int n_in,
                              void* d_out, int out_size, void* d_ws, size_t ws_size,
                              hipStream_t stream)
{
    (void)n_in; (void)d_ws; (void)ws_size; (void)out_size;
    const float* x  = (const float*)d_in[0];   // [B, L, P] fp32
    const float* w  = (const float*)d_in[1];   // [P] fp32
    float*       o  = (float*)d_out;

    const int total = in_sizes[0];             // B*L*P = 122,683,392
    const int rows  = total / P;               // 9,437,184 (exact)
    const int grid  = rows / ROWS_PER_BLOCK;   // 36,864 (exact for B=65536, L=144)

    chordal_pc_softmax_kernel<<<grid, BLOCK, 0, stream>>>(x, w, o);
}